// FastIntegralKernel_23751169147525
// MI455X (gfx1250) — compile-verified
//
#include <hip/hip_runtime.h>

typedef __attribute__((ext_vector_type(2))) float v2f;
typedef __attribute__((ext_vector_type(8))) float v8f;

#define B_      8
#define N_      1048576
#define Z_      1024
#define HID_    16
#define THREADS 256
#define WAVES   8
#define EPB     4096                 // elements per block (divides N_)
#define EPW     (EPB / WAVES)        // 512 elements per wave
#define ITERS   (EPW / 16)           // 32 WMMA steps per wave
#define BLOCKS  ((B_ * N_) / EPB)    // 2048 blocks
#define BLOCKS_PER_BATCH (N_ / EPB)  // 256
#define PFD     256                  // prefetch distance (elements, = 16 iters)

// xor-16 lane swap as pure VALU (v_permlanex16_b32 with identity selects):
// lane i (0-15) <-> lane 16+i. No DS round-trip.
__device__ __forceinline__ float xor16(float v) {
    const unsigned u = __float_as_uint(v);
    const unsigned r = __builtin_amdgcn_permlanex16(
        u, u, 0x76543210u, 0xfedcba98u, false, false);
    return __uint_as_float(r);
}

// Branch-free erf-based GELU (Abramowitz-Stegun 7.1.26, |err| <= 1.5e-7).
// v_rcp_f32 + v_exp_f32 + FMAs only; no EXEC divergence.
__device__ __forceinline__ float gelu_erf(float h) {
    const float u  = 0.70710678118654752f * h;
    const float xa = fabsf(u);
    const float t  = __builtin_amdgcn_rcpf(fmaf(0.3275911f, xa, 1.0f));
    float p = fmaf(1.061405429f, t, -1.453152027f);
    p = fmaf(p, t, 1.421413741f);
    p = fmaf(p, t, -0.284496736f);
    p = fmaf(p, t, 0.254829592f);
    p = p * t;
    const float ea   = fmaf(-p, __expf(-u * u), 1.0f);  // erf(|u|)
    const float erfu = copysignf(ea, u);
    return 0.5f * h * (1.0f + erfu);
}

__global__ __launch_bounds__(THREADS) void fik_main(
    const float* __restrict__ x,  const float* __restrict__ y,
    const float* __restrict__ z,  const float* __restrict__ W1,
    const float* __restrict__ b1, const float* __restrict__ gamma,
    const float* __restrict__ beta, const float* __restrict__ W2,
    const float* __restrict__ b2, float* __restrict__ gsums,
    float* __restrict__ gcnts)
{
    __shared__ float lsum[Z_];
    __shared__ float lcnt[Z_];

    const int tid = threadIdx.x;
    for (int i = tid; i < Z_; i += THREADS) { lsum[i] = 0.0f; lcnt[i] = 0.0f; }
    __syncthreads();

    const int  lane = tid & 31;
    const int  wave = tid >> 5;
    const bool hi   = (lane >= 16);
    const int  l16  = lane & 15;
    const int  moff = hi ? 8 : 0;

    // Loop-invariant A fragment: A = [W1^T | b1], 16x4 (M=hidden, K=feature).
    // Lane L<16 holds A[L][0..1] in v0,v1; lane L+16 holds A[L][2..3].
    v2f a;
    if (!hi) { a.x = W1[0 * HID_ + l16]; a.y = W1[1 * HID_ + l16]; }
    else     { a.x = W1[2 * HID_ + l16]; a.y = b1[l16]; }

    // Per-lane slices of gamma/beta/W2 for rows m = moff..moff+7.
    float g[8], bt[8], w2[8];
#pragma unroll
    for (int i = 0; i < 8; ++i) {
        g[i]  = gamma[moff + i];
        bt[i] = beta[moff + i];
        w2[i] = W2[moff + i];
    }
    const float b2s = b2[0];
    const float z0  = z[0];
    const float dz  = z[1] - z0;
    const float inv_dz = 1.0f / dz;

    const long long base = (long long)blockIdx.x * EPB + (long long)wave * EPW;

#pragma unroll 2
    for (int t = 0; t < ITERS; ++t) {
        const long long e = base + t * 16 + l16;   // lane-pair n / n+16 share element
        const float xv = x[e];
        const float yv = y[e];
        __builtin_prefetch(&x[e + PFD], 0, 3);     // global_prefetch_b8
        __builtin_prefetch(&y[e + PFD], 0, 3);

        // Bucket index & its z value, clamped in float domain (one cvt only).
        float idxf = ceilf((xv - z0 - 0.5f * dz) * inv_dz);
        idxf = fminf(fmaxf(idxf, 0.0f), (float)(Z_ - 1));
        const int   idx = (int)idxf;
        const float zg  = fmaf(idxf, dz, z0);      // == z[idx] (linspace), no gather

        // B fragment: kin^T (4x16). K rows: 0=x, 1=zg, 2=y, 3=1 (bias row).
        v2f bf;
        bf.x = hi ? yv   : xv;   // v0: K=0 (lanes 0-15), K=2 (lanes 16-31)
        bf.y = hi ? 1.0f : zg;   // v1: K=1 (lanes 0-15), K=3 (lanes 16-31)

        v8f c = {};
        // Full Linear1 + bias for 16 elements in one matrix op.
        v8f d = __builtin_amdgcn_wmma_f32_16x16x4_f32(
            false, a, false, bf, (short)0, c, false, false);

        // LayerNorm reduction over adjacent VGPR pairs (packed, no movs),
        // then one xor16 VALU swap to combine the two half-columns.
        const v2f p0 = {d[0], d[1]};
        const v2f p1 = {d[2], d[3]};
        const v2f p2 = {d[4], d[5]};
        const v2f p3 = {d[6], d[7]};
        const v2f sv = (p0 + p1) + (p2 + p3);
        v2f qv = p0 * p0;
        qv = {fmaf(p1.x, p1.x, qv.x), fmaf(p1.y, p1.y, qv.y)};
        qv = {fmaf(p2.x, p2.x, qv.x), fmaf(p2.y, p2.y, qv.y)};
        qv = {fmaf(p3.x, p3.x, qv.x), fmaf(p3.y, p3.y, qv.y)};
        float s  = sv.x + sv.y;
        float s2 = qv.x + qv.y;
        s  += xor16(s);
        s2 += xor16(s2);
        const float mu   = s * (1.0f / 16.0f);
        const float var  = s2 * (1.0f / 16.0f) - mu * mu;
        // var + 1e-5 >= 1e-5: far above denorms, raw v_rsq_f32 is safe.
        const float rstd = __builtin_amdgcn_rsqf(var + 1e-5f);

        // Normalize, affine, GELU(erf), Linear2 partial dot.
        float p = 0.0f;
#pragma unroll
        for (int i = 0; i < 8; ++i) {
            const float h = fmaf((d[i] - mu) * rstd, g[i], bt[i]);
            p = fmaf(gelu_erf(h), w2[i], p);
        }
        p += xor16(p);
        const float out = (p + b2s) * yv;

        // One full-EXEC ds_add_f32: low half commits the sum, high half the count.
        float* aptr = hi ? &lcnt[idx] : &lsum[idx];
        atomicAdd(aptr, hi ? 1.0f : out);
    }

    __syncthreads();
    const int b = blockIdx.x / BLOCKS_PER_BATCH;   // block never crosses a batch
    for (int i = tid; i < Z_; i += THREADS) {
        atomicAdd(&gsums[b * Z_ + i], lsum[i]);
        atomicAdd(&gcnts[b * Z_ + i], lcnt[i]);
    }
}

__global__ void fik_zero(float* __restrict__ ws, int n) {
    const int i = blockIdx.x * 256 + threadIdx.x;
    if (i < n) ws[i] = 0.0f;
}

__global__ void fik_final(const float* __restrict__ gsums,
                          const float* __restrict__ gcnts,
                          float* __restrict__ out) {
    const int i = blockIdx.x * 256 + threadIdx.x;
    if (i < B_ * Z_) out[i] = gsums[i] / fmaxf(gcnts[i], 1.0f);
}

extern "C" void kernel_launch(void* const* d_in, const int* in_sizes, int n_in,
                              void* d_out, int out_size, void* d_ws, size_t ws_size,
                              hipStream_t stream) {
    (void)in_sizes; (void)n_in; (void)out_size; (void)ws_size;
    const float* x     = (const float*)d_in[0];
    const float* y     = (const float*)d_in[1];
    const float* z     = (const float*)d_in[2];
    const float* W1    = (const float*)d_in[3];
    const float* b1    = (const float*)d_in[4];
    const float* gamma = (const float*)d_in[5];
    const float* beta  = (const float*)d_in[6];
    const float* W2    = (const float*)d_in[7];
    const float* b2    = (const float*)d_in[8];

    float* gsums = (float*)d_ws;            // B_*Z_ floats
    float* gcnts = gsums + B_ * Z_;         // B_*Z_ floats

    const int accN = 2 * B_ * Z_;
    fik_zero<<<(accN + 255) / 256, 256, 0, stream>>>(gsums, accN);

    fik_main<<<BLOCKS, THREADS, 0, stream>>>(x, y, z, W1, b1, gamma, beta,
                                             W2, b2, gsums, gcnts);

    fik_final<<<(B_ * Z_ + 255) / 256, 256, 0, stream>>>(gsums, gcnts,
                                                         (float*)d_out);
}